// URNN_34686155882833
// MI455X (gfx1250) — compile-verified
//
#include <hip/hip_runtime.h>
#include <math.h>

// Problem sizes (match reference)
#define BATCH  2048
#define INPUT  128
#define HID    2048
#define FIVEH  (5 * HID)
#define NTH    256

typedef __attribute__((ext_vector_type(2))) float v2f;
typedef __attribute__((ext_vector_type(8))) float v8f;

// ---------------------------------------------------------------------------
// Kernel 1: up = input @ W^T + bias, fp32 WMMA (V_WMMA_F32_16X16X4_F32).
// Grid: (BATCH/32, FIVEH/64), block 256 threads = 8 waves, each wave a 16x16
// C tile; K = 128 -> 32 wmma ops per wave.
// ---------------------------------------------------------------------------
__launch_bounds__(NTH)
__global__ void urnn_gemm_wmma(const float* __restrict__ A,     // (BATCH, 128)
                               const float* __restrict__ W,     // (FIVEH, 128)
                               const float* __restrict__ bias,  // (FIVEH)
                               float* __restrict__ up) {        // (BATCH, FIVEH)
  const int lane = threadIdx.x & 31;
  const int wave = threadIdx.x >> 5;
  const int m0 = blockIdx.x * 32 + (wave >> 2) * 16;
  const int n0 = blockIdx.y * 64 + (wave & 3) * 16;
  const int r  = lane & 15;          // A row / B col within tile
  const int kk = (lane >> 4) * 2;    // K sub-offset within fragment

  const float* __restrict__ arow = A + (size_t)(m0 + r) * INPUT + kk;
  const float* __restrict__ brow = W + (size_t)(n0 + r) * INPUT + kk;

  // Prefetch the back half of both K-streams (lowers to global_prefetch_b8).
  __builtin_prefetch(arow + 64, 0, 3);
  __builtin_prefetch(brow + 64, 0, 3);

  v8f c = {};
#pragma unroll
  for (int k0 = 0; k0 < INPUT; k0 += 4) {
    v2f a;
    a.x = arow[k0];
    a.y = arow[k0 + 1];
    v2f b;
    b.x = brow[k0];
    b.y = brow[k0 + 1];
    // (neg_a, A, neg_b, B, c_mod, C, reuse_a, reuse_b)
    c = __builtin_amdgcn_wmma_f32_16x16x4_f32(false, a, false, b, (short)0, c,
                                              false, false);
  }

  const float bv = bias[n0 + r];
  const int mbase = m0 + (lane >> 4) * 8;
#pragma unroll
  for (int i = 0; i < 8; ++i) {
    up[(size_t)(mbase + i) * FIVEH + (n0 + r)] = c[i] + bv;
  }
}

// ---------------------------------------------------------------------------
// Kernel 2: fully fused per-row URNN pipeline, one block per batch row.
//   async-stage hx row to LDS (overlapped with sincos(d1)) -> t = d1*hx
//   -> FFT(2048) -> Householder(r1) -> permute*d2 -> IFFT(2048)
//   -> Householder(r2) -> *d3 -> ModReLU -> out
// The row lives in LDS the whole time (2 x 16KB ping-pong buffers).
// ---------------------------------------------------------------------------
__device__ __forceinline__ void block_reduce2(float2* red, int tid, float2 acc,
                                              float2* out) {
  red[tid] = acc;
  __syncthreads();
  for (int off = NTH >> 1; off > 0; off >>= 1) {
    if (tid < off) {
      red[tid].x += red[tid + off].x;
      red[tid].y += red[tid + off].y;
    }
    __syncthreads();
  }
  *out = red[0];
}

__launch_bounds__(NTH)
__global__ void urnn_fused_kernel(const float* __restrict__ hxr,
                                  const float* __restrict__ hxi,
                                  const float* __restrict__ up,
                                  const float* __restrict__ beta,
                                  const int* __restrict__ perm,
                                  float2* __restrict__ out) {
  __shared__ __attribute__((aligned(16))) float2 shA[HID];
  __shared__ __attribute__((aligned(16))) float2 shB[HID];
  __shared__ float2 red[NTH];

  const int b = blockIdx.x;
  const int tid = threadIdx.x;
  const float* __restrict__ uprow = up + (size_t)b * FIVEH;
  const float* __restrict__ d1 = uprow;
  const float* __restrict__ d2 = uprow + HID;
  const float* __restrict__ d3 = uprow + 2 * HID;
  const float* __restrict__ r1 = uprow + 3 * HID;
  const float* __restrict__ r2 = uprow + 4 * HID;

  // ---- async-stage hx row into LDS (GLOBAL_LOAD_ASYNC_TO_LDS_B128) ----
  // Layout in shB (reused as raw float staging): [0,2048) = hx_real row,
  // [2048,4096) = hx_imag row. 2048 floats / 256 threads = 2 x b128 each.
  {
    const unsigned ldsbase = (unsigned)(uintptr_t)(float*)shB;
    const unsigned long long gr =
        (unsigned long long)(uintptr_t)(hxr + (size_t)b * HID);
    const unsigned long long gi =
        (unsigned long long)(uintptr_t)(hxi + (size_t)b * HID);
#pragma unroll
    for (int e = 0; e < 2; ++e) {
      const unsigned foff = (unsigned)((tid + e * NTH) * 4);  // float index
      const unsigned gb = foff * 4u;                          // byte offset
      const unsigned lr = ldsbase + foff * 4u;
      const unsigned li = ldsbase + (2048u + foff) * 4u;
      asm volatile("global_load_async_to_lds_b128 %0, %1, %2"
                   :: "v"(lr), "v"(gb), "s"(gr) : "memory");
      asm volatile("global_load_async_to_lds_b128 %0, %1, %2"
                   :: "v"(li), "v"(gb), "s"(gi) : "memory");
    }
  }

  // ---- compute phases exp(i*d1) while the async DMA fills LDS ----
  float pc[8], ps[8];
#pragma unroll
  for (int e = 0; e < 8; ++e) {
    __sincosf(d1[tid + e * NTH], &ps[e], &pc[e]);
  }

  asm volatile("s_wait_asynccnt 0x0" ::: "memory");
  __syncthreads();

  // ---- t = d1 * hx ----
  {
    const float* hr_st = (const float*)shB;
    const float* hi_st = hr_st + 2048;
#pragma unroll
    for (int e = 0; e < 8; ++e) {
      const int k = tid + e * NTH;
      const float xr = hr_st[k];
      const float xi = hi_st[k];
      shA[k] = make_float2(pc[e] * xr - ps[e] * xi, pc[e] * xi + ps[e] * xr);
    }
  }
  __syncthreads();

  float2* src = shA;
  float2* dst = shB;

  // ---- forward FFT, radix-2 Stockham, 11 stages ----
  for (int Ns = 1; Ns < HID; Ns <<= 1) {
    for (int i = tid; i < HID / 2; i += NTH) {
      const int j = i % Ns;
      float ws_, wc_;
      __sincosf(-3.14159265358979323846f * (float)j / (float)Ns, &ws_, &wc_);
      const float2 a = src[i];
      const float2 t = src[i + HID / 2];
      const float2 bt =
          make_float2(wc_ * t.x - ws_ * t.y, wc_ * t.y + ws_ * t.x);
      const int o = ((i - j) << 1) + j;  // (i/Ns)*2Ns + j
      dst[o] = make_float2(a.x + bt.x, a.y + bt.y);
      dst[o + Ns] = make_float2(a.x - bt.x, a.y - bt.y);
    }
    __syncthreads();
    float2* tmp = src; src = dst; dst = tmp;
  }

  // ---- Householder with v = exp(i*r1): h -= 2 v (v^H h) ----
  {
    float2 acc = make_float2(0.f, 0.f);
    for (int k = tid; k < HID; k += NTH) {
      float s, c;
      __sincosf(r1[k], &s, &c);
      const float2 h = src[k];
      acc.x += c * h.x + s * h.y;   // Re(conj(v)*h)
      acc.y += c * h.y - s * h.x;   // Im(conj(v)*h)
    }
    float2 proj;
    block_reduce2(red, tid, acc, &proj);
    for (int k = tid; k < HID; k += NTH) {
      float s, c;
      __sincosf(r1[k], &s, &c);
      const float2 h = src[k];
      const float pr = c * proj.x - s * proj.y;
      const float pi = c * proj.y + s * proj.x;
      src[k] = make_float2(h.x - 2.f * pr, h.y - 2.f * pi);
    }
    __syncthreads();
  }

  // ---- g = d2 * h[perm] ----
  for (int k = tid; k < HID; k += NTH) {
    float s, c;
    __sincosf(d2[k], &s, &c);
    const float2 h = src[perm[k]];
    dst[k] = make_float2(c * h.x - s * h.y, c * h.y + s * h.x);
  }
  __syncthreads();
  { float2* tmp = src; src = dst; dst = tmp; }

  // ---- inverse FFT ----
  for (int Ns = 1; Ns < HID; Ns <<= 1) {
    for (int i = tid; i < HID / 2; i += NTH) {
      const int j = i % Ns;
      float ws_, wc_;
      __sincosf(3.14159265358979323846f * (float)j / (float)Ns, &ws_, &wc_);
      const float2 a = src[i];
      const float2 t = src[i + HID / 2];
      const float2 bt =
          make_float2(wc_ * t.x - ws_ * t.y, wc_ * t.y + ws_ * t.x);
      const int o = ((i - j) << 1) + j;
      dst[o] = make_float2(a.x + bt.x, a.y + bt.y);
      dst[o + Ns] = make_float2(a.x - bt.x, a.y - bt.y);
    }
    __syncthreads();
    float2* tmp = src; src = dst; dst = tmp;
  }

  // ---- 1/N scaling for the inverse transform ----
  const float invH = 1.0f / (float)HID;
  for (int k = tid; k < HID; k += NTH) {
    src[k].x *= invH;
    src[k].y *= invH;
  }
  __syncthreads();

  // ---- Householder with v = exp(i*r2), fused with d3 and ModReLU ----
  {
    float2 acc = make_float2(0.f, 0.f);
    for (int k = tid; k < HID; k += NTH) {
      float s, c;
      __sincosf(r2[k], &s, &c);
      const float2 h = src[k];
      acc.x += c * h.x + s * h.y;
      acc.y += c * h.y - s * h.x;
    }
    float2 proj;
    block_reduce2(red, tid, acc, &proj);
    for (int k = tid; k < HID; k += NTH) {
      float s, c;
      __sincosf(r2[k], &s, &c);
      const float2 h = src[k];
      const float pr = c * proj.x - s * proj.y;
      const float pi = c * proj.y + s * proj.x;
      const float hr0 = h.x - 2.f * pr;
      const float hi0 = h.y - 2.f * pi;
      // h = d3 * h
      float s3, c3;
      __sincosf(d3[k], &s3, &c3);
      const float hr = c3 * hr0 - s3 * hi0;
      const float hi = c3 * hi0 + s3 * hr0;
      // ModReLU
      const float mag = sqrtf(hr * hr + hi * hi);
      const float nm = fmaxf(mag + beta[k], 0.0f);
      float2 o;
      if (mag > 0.0f) {
        const float sc = nm / mag;
        o = make_float2(hr * sc, hi * sc);
      } else {
        o = make_float2(nm, 0.0f);  // angle(0) == 0 -> exp(i*0) == 1
      }
      out[(size_t)b * HID + k] = o;
    }
  }
}

// ---------------------------------------------------------------------------
extern "C" void kernel_launch(void* const* d_in, const int* in_sizes, int n_in,
                              void* d_out, int out_size, void* d_ws,
                              size_t ws_size, hipStream_t stream) {
  (void)in_sizes; (void)n_in; (void)out_size; (void)ws_size;

  const float* input = (const float*)d_in[0];  // (BATCH, INPUT)
  const float* hxr   = (const float*)d_in[1];  // (BATCH, HID)
  const float* hxi   = (const float*)d_in[2];  // (BATCH, HID)
  const float* W     = (const float*)d_in[3];  // (FIVEH, INPUT)
  const float* bu    = (const float*)d_in[4];  // (FIVEH)
  const float* beta  = (const float*)d_in[5];  // (HID)
  const int*   perm  = (const int*)d_in[6];    // (HID)

  float* up = (float*)d_ws;  // (BATCH, FIVEH) fp32 scratch = 80 MB

  dim3 gGemm(BATCH / 32, FIVEH / 64);
  urnn_gemm_wmma<<<gGemm, NTH, 0, stream>>>(input, W, bu, up);

  urnn_fused_kernel<<<BATCH, NTH, 0, stream>>>(hxr, hxi, up, beta, perm,
                                               (float2*)d_out);
}